// MultiSensor_TimeMIL_v2_670014898678
// MI455X (gfx1250) — compile-verified
//
// MI455X (gfx1250) implementation of the MultiSensor_TimeMIL forward pass.
//
// Roofline: ~0.2 TFLOP fp32, hot tensors L2-resident (192MB L2, 23.3TB/s HBM)
// -> matrix-pipe bound. All GEMMs (projections, 3x MHA, qkv, Nystrom a1/a2/a3,
// 6x Moore-Penrose Newton iterations = ~100 GFLOP of batched 256^3 GEMMs, out
// projections, classifier heads) run through V_WMMA_F32_16X16X4_F32 (CDNA5's
// native fp32 matrix op, wave32, 2048 FLOP/inst/wave). fp32 kept end-to-end to
// match the reference (pinv Newton iteration is precision-sensitive).
//
// Round-2 change: branch-free GEMM inner loop. OOB rows/cols are CLAMPED (the
// garbage they compute is never stored) and the K remainder is peeled, so the
// hot loop is 8 unconditional global_load_b32 + 4 v_wmma per k-step instead of
// 8 saveexec/branch/reconverge sequences.
//
// Input flattening assumption: setup_inputs() dict order, depth-first
// insertion order -> 84 flat device pointers (documented per index below).

#include <hip/hip_runtime.h>
#include <math.h>

typedef float v2f __attribute__((ext_vector_type(2)));
typedef float v8f __attribute__((ext_vector_type(8)));

#ifndef __has_builtin
#define __has_builtin(x) 0
#endif
#if __has_builtin(__builtin_amdgcn_wmma_f32_16x16x4_f32)
#define HAVE_WMMA_F32 1
#else
#define HAVE_WMMA_F32 0
#endif

static __device__ __forceinline__ v8f wmma4(v2f a, v2f b, v8f c) {
#if HAVE_WMMA_F32
  // (neg_a, A, neg_b, B, c_mod, C, reuse_a, reuse_b)
  return __builtin_amdgcn_wmma_f32_16x16x4_f32(false, a, false, b, (short)0, c,
                                               false, false);
#else
  c[0] += a.x * b.x + a.y * b.y;  // placeholder: histogram would show wmma=0
  return c;
#endif
}

// ---------------------------------------------------------------------------
// Generic batched GEMM: C = alpha*(A @ opB(B)) + bias + diag*I + resid, (relu)
// Two-level batch: z = zo*inner + zi, operand offset = zo*s?o + zi*s?i.
// Wave computes a 32x32 tile (2x2 fp32 WMMA frags); 8 waves/WG cover 64x128.
// Fragment layouts per CDNA5 ISA 7.12.2: reg v, lane-half g -> K/M = v + (n/2)g.
// ---------------------------------------------------------------------------
template <int TB>
__global__ __launch_bounds__(256) void gemm_wmma(
    const float* __restrict__ A, int lda, long long sAo, long long sAi,
    const float* __restrict__ Bm, int ldb, long long sBo, long long sBi,
    float* __restrict__ C, int ldc, long long sCo, long long sCi,
    const float* __restrict__ bias, const float* __restrict__ resid, int M,
    int N, int K, int inner, float alpha, float diag, int relu) {
  int z = blockIdx.z, zo = z / inner, zi = z - zo * inner;
  const float* Ab = A + zo * sAo + zi * sAi;
  const float* Bb = Bm + zo * sBo + zi * sBi;
  float* Cb = C + zo * sCo + zi * sCi;
  const float* Rb = resid ? resid + zo * sCo + zi * sCi : (const float*)0;

  int wave = threadIdx.x >> 5, lane = threadIdx.x & 31;
  int wr = wave >> 2, wc = wave & 3;  // 2 x 4 waves
  int m0 = blockIdx.y * 64 + wr * 32;
  int n0 = blockIdx.x * 128 + wc * 32;
  if (m0 >= M || n0 >= N) return;  // wave-uniform: keeps EXEC all-1 for WMMA

  int half = lane >> 4, l16 = lane & 15;
  int kofs = half ? 2 : 0;  // frag element 0 holds K = k + 2*half (ISA 7.12.2)
  int ra0 = m0 + l16, ra1 = m0 + 16 + l16;
  int cb0 = n0 + l16, cb1 = n0 + 16 + l16;
  bool vA0 = ra0 < M, vA1 = ra1 < M, vB0 = cb0 < N, vB1 = cb1 < N;
  // Clamp OOB lanes to a valid row/col: their results are never stored, and
  // this keeps the inner loop free of exec-mask manipulation.
  int ra0c = vA0 ? ra0 : M - 1;
  int ra1c = vA1 ? ra1 : M - 1;
  int cb0c = vB0 ? cb0 : N - 1;
  int cb1c = vB1 ? cb1 : N - 1;

  const float* Ar0 = Ab + (long long)ra0c * lda + kofs;
  const float* Ar1 = Ab + (long long)ra1c * lda + kofs;
  const float* Bc0;
  const float* Bc1;
  long long bstep;
  if (TB) {  // B[n][k]
    Bc0 = Bb + (long long)cb0c * ldb + kofs;
    Bc1 = Bb + (long long)cb1c * ldb + kofs;
    bstep = 4;
  } else {  // B[k][n]
    Bc0 = Bb + cb0c + (long long)kofs * ldb;
    Bc1 = Bb + cb1c + (long long)kofs * ldb;
    bstep = 4ll * ldb;
  }

  v8f z8 = {0.f, 0.f, 0.f, 0.f, 0.f, 0.f, 0.f, 0.f};
  v8f acc00 = z8, acc01 = z8, acc10 = z8, acc11 = z8;

  int Kmain = K & ~3;
#pragma unroll 2
  for (int k = 0; k < Kmain; k += 4) {
    v2f a0, a1, b0, b1;
    a0.x = Ar0[0];
    a0.y = Ar0[1];
    a1.x = Ar1[0];
    a1.y = Ar1[1];
    if (TB) {
      b0.x = Bc0[0];
      b0.y = Bc0[1];
      b1.x = Bc1[0];
      b1.y = Bc1[1];
    } else {
      b0.x = Bc0[0];
      b0.y = Bc0[ldb];
      b1.x = Bc1[0];
      b1.y = Bc1[ldb];
    }
    Ar0 += 4;
    Ar1 += 4;
    Bc0 += bstep;
    Bc1 += bstep;
    acc00 = wmma4(a0, b0, acc00);
    acc01 = wmma4(a0, b1, acc01);
    acc10 = wmma4(a1, b0, acc10);
    acc11 = wmma4(a1, b1, acc11);
  }
  if (K & 3) {  // peeled, guarded tail (only K=511 paths take this)
    int k0 = Kmain + kofs, k1 = k0 + 1;
    bool v0 = k0 < K, v1 = k1 < K;
    v2f a0, a1, b0, b1;
    a0.x = v0 ? Ar0[0] : 0.f;
    a0.y = v1 ? Ar0[1] : 0.f;
    a1.x = v0 ? Ar1[0] : 0.f;
    a1.y = v1 ? Ar1[1] : 0.f;
    if (TB) {
      b0.x = v0 ? Bc0[0] : 0.f;
      b0.y = v1 ? Bc0[1] : 0.f;
      b1.x = v0 ? Bc1[0] : 0.f;
      b1.y = v1 ? Bc1[1] : 0.f;
    } else {
      b0.x = v0 ? Bc0[0] : 0.f;
      b0.y = v1 ? Bc0[ldb] : 0.f;
      b1.x = v0 ? Bc1[0] : 0.f;
      b1.y = v1 ? Bc1[ldb] : 0.f;
    }
    acc00 = wmma4(a0, b0, acc00);
    acc01 = wmma4(a0, b1, acc01);
    acc10 = wmma4(a1, b0, acc10);
    acc11 = wmma4(a1, b1, acc11);
  }

  auto store_tile = [&](v8f acc, int tm, int tn) {
    int col = tn + l16;
    if (col >= N) return;
    float bv = bias ? bias[col] : 0.f;
#pragma unroll
    for (int r = 0; r < 8; ++r) {
      int row = tm + r + (half ? 8 : 0);
      if (row >= M) continue;
      float vv = alpha * acc[r] + bv;
      if (diag != 0.f && row == col) vv += diag;
      long long ci = (long long)row * ldc + col;
      if (Rb) vv += Rb[ci];
      if (relu) vv = fmaxf(vv, 0.f);
      Cb[ci] = vv;
    }
  };
  store_tile(acc00, m0, n0);
  store_tile(acc01, m0, n0 + 16);
  store_tile(acc10, m0 + 16, n0);
  store_tile(acc11, m0 + 16, n0 + 16);
}

// ---------------------------------------------------------------------------
// Elementwise / reduction kernels
// ---------------------------------------------------------------------------

// feature-extraction conv1d (K=9, pad 4) + bias + relu, writes concat layout
__global__ void fe_conv_relu(const float* __restrict__ x,
                             const float* __restrict__ w,
                             const float* __restrict__ bias,
                             float* __restrict__ out, int B, int S, int L,
                             int Cin, int OCrow) {
  int idx = blockIdx.x * blockDim.x + threadIdx.x;
  int total = B * L * S * 128;
  if (idx >= total) return;
  int oc = idx & 127;
  int s = (idx >> 7) % S;
  int r = (idx >> 7) / S;
  int l = r % L;
  int b = r / L;
  float acc = bias[oc];
  const float* wrow = w + (size_t)oc * Cin * 9;
  const float* xb = x + ((size_t)(b * S + s) * L) * Cin;
  for (int c = 0; c < Cin; ++c)
    for (int t = 0; t < 9; ++t) {
      int l2 = l + t - 4;
      if (l2 >= 0 && l2 < L) acc += xb[(size_t)l2 * Cin + c] * wrow[c * 9 + t];
    }
  out[((size_t)(b * L + l)) * OCrow + s * 128 + oc] = fmaxf(acc, 0.f);
}

// sum of 3 mexican-hat kernels (conv is linear -> one 19-tap depthwise kernel)
__global__ void build_wavelet(const float* __restrict__ w1,
                              const float* __restrict__ w2,
                              const float* __restrict__ w3,
                              float* __restrict__ wk) {
  int idx = blockIdx.x * blockDim.x + threadIdx.x;
  if (idx >= 512 * 19) return;
  int d = idx / 19, j = idx % 19;
  float xj = (float)(j - 9);
  const float* ws[3] = {w1, w2, w3};
  float s = 0.f;
  for (int i = 0; i < 3; ++i) {
    float scale = ws[i][d], shift = ws[i][512 + d];
    float u = (xj - shift) / scale;
    // C = 2 / (3^0.5 * pi^0.25)
    s += 0.86732507f * (1.f - u * u) * __expf(-0.5f * u * u) *
         rsqrtf(fabsf(scale));
  }
  wk[idx] = s;
}

// depthwise 19-tap conv over tokens 1..510 of x (B,511,512) -> pos (B,510,512)
__global__ void wavelet_pos(const float* __restrict__ x,
                            const float* __restrict__ wk,
                            float* __restrict__ pos, int B) {
  int idx = blockIdx.x * blockDim.x + threadIdx.x;
  int total = B * 510 * 512;
  if (idx >= total) return;
  int d = idx & 511;
  int r = idx >> 9;
  int t = r % 510;
  int b = r / 510;
  const float* xb = x + ((size_t)b * 511 + 1) * 512 + d;
  const float* kd = wk + d * 19;
  float s = 0.f;
#pragma unroll
  for (int j = 0; j < 19; ++j) {
    int tt = t + j - 9;
    if (tt >= 0 && tt < 510) s += xb[(size_t)tt * 512] * kd[j];
  }
  pos[(size_t)idx] = s;
}

// LayerNorm over D=512, wave-per-row (wave32: 16 elems/lane), optional resid
__global__ __launch_bounds__(256) void layer_norm_k(
    const float* __restrict__ x, const float* __restrict__ resid,
    const float* __restrict__ g, const float* __restrict__ bprm,
    float* __restrict__ y, int rows, int rpb, long long xbs, long long xrs,
    long long ybs, long long yrs) {
  int row = blockIdx.x * (blockDim.x >> 5) + (threadIdx.x >> 5);
  if (row >= rows) return;
  int lane = threadIdx.x & 31;
  int bb = row / rpb, rr = row - bb * rpb;
  const float* xr = x + bb * xbs + rr * xrs;
  const float* rp = resid ? resid + bb * xbs + rr * xrs : (const float*)0;
  float* yr = y + bb * ybs + rr * yrs;
  float v[16];
  float s = 0.f, s2 = 0.f;
#pragma unroll
  for (int i = 0; i < 16; ++i) {
    int idx = lane + i * 32;
    float t = xr[idx] + (rp ? rp[idx] : 0.f);
    v[i] = t;
    s += t;
    s2 += t * t;
  }
  for (int off = 16; off; off >>= 1) {
    s += __shfl_xor(s, off);
    s2 += __shfl_xor(s2, off);
  }
  float mean = s * (1.f / 512.f);
  float var = s2 * (1.f / 512.f) - mean * mean;
  float rinv = rsqrtf(var + 1e-5f);
#pragma unroll
  for (int i = 0; i < 16; ++i) {
    int idx = lane + i * 32;
    yr[idx] = (v[i] - mean) * rinv * g[idx] + bprm[idx];
  }
}

// in-place row softmax, wave-per-row
__global__ __launch_bounds__(256) void softmax_rows(float* __restrict__ x,
                                                    long long rows, int len) {
  long long row =
      (long long)blockIdx.x * (blockDim.x >> 5) + (threadIdx.x >> 5);
  if (row >= rows) return;
  int lane = threadIdx.x & 31;
  float* p = x + row * (long long)len;
  float m = -3.4e38f;
  for (int i = lane; i < len; i += 32) m = fmaxf(m, p[i]);
  for (int off = 16; off; off >>= 1) m = fmaxf(m, __shfl_xor(m, off));
  float s = 0.f;
  for (int i = lane; i < len; i += 32) {
    float e = __expf(p[i] - m);
    p[i] = e;
    s += e;
  }
  for (int off = 16; off; off >>= 1) s += __shfl_xor(s, off);
  float r = 1.f / s;
  for (int i = lane; i < len; i += 32) p[i] *= r;
}

// landmark means: out[(b,h),j,d] = 0.5*(sec[b,2j,h,d] + sec[b,2j+1,h,d])
__global__ void landmark_mean(const float* __restrict__ qkv,
                              float* __restrict__ out, int sec) {
  int idx = blockIdx.x * blockDim.x + threadIdx.x;
  if (idx >= 16 * 8 * 256 * 64) return;
  int d = idx & 63;
  int j = (idx >> 6) & 255;
  int h = (idx >> 14) & 7;
  int b = idx >> 17;
  const float* base = qkv + ((size_t)b * 512 + 2 * j) * 1536 + sec + h * 64 + d;
  out[idx] = 0.5f * (base[0] + base[1536]);
}

__global__ void abs_rowsum(const float* __restrict__ a,
                           float* __restrict__ out) {
  int r = blockIdx.x * blockDim.x + threadIdx.x;
  if (r >= 32768) return;
  const float* p = a + (size_t)r * 256;
  float s = 0.f;
  for (int t = 0; t < 256; ++t) s += fabsf(p[t]);
  out[r] = s;
}

__global__ void abs_colsum(const float* __restrict__ a,
                           float* __restrict__ out) {
  int r = blockIdx.x * blockDim.x + threadIdx.x;
  if (r >= 32768) return;
  int bh = r >> 8, j = r & 255;
  const float* p = a + ((size_t)bh << 16) + j;
  float s = 0.f;
  for (int t = 0; t < 256; ++t) s += fabsf(p[t << 8]);
  out[r] = s;
}

__global__ void pinv_scale(const float* __restrict__ R,
                           const float* __restrict__ Cc, float* __restrict__ sc,
                           int n) {
  __shared__ float sR[256], sC[256];
  float mr = 0.f, mc = 0.f;
  for (int i = threadIdx.x; i < n; i += 256) {
    mr = fmaxf(mr, R[i]);
    mc = fmaxf(mc, Cc[i]);
  }
  sR[threadIdx.x] = mr;
  sC[threadIdx.x] = mc;
  __syncthreads();
  for (int off = 128; off; off >>= 1) {
    if ((int)threadIdx.x < off) {
      sR[threadIdx.x] = fmaxf(sR[threadIdx.x], sR[threadIdx.x + off]);
      sC[threadIdx.x] = fmaxf(sC[threadIdx.x], sC[threadIdx.x + off]);
    }
    __syncthreads();
  }
  if (threadIdx.x == 0) sc[0] = 1.f / (sR[0] * sC[0]);
}

// z0 = a^T * scale (batched 256x256)
__global__ void transpose_scale(const float* __restrict__ a,
                                const float* __restrict__ sc,
                                float* __restrict__ z) {
  int idx = blockIdx.x * blockDim.x + threadIdx.x;
  if (idx >= 128 * 256 * 256) return;
  int j = idx & 255, i = (idx >> 8) & 255, bh = idx >> 16;
  z[idx] = a[((size_t)bh << 16) + ((size_t)j << 8) + i] * sc[0];
}

// T = c*I - X (batched 256x256)
__global__ void ident_sub(const float* __restrict__ x, float* __restrict__ t,
                          float c) {
  int idx = blockIdx.x * blockDim.x + threadIdx.x;
  if (idx >= 128 * 256 * 256) return;
  int j = idx & 255, i = (idx >> 8) & 255;
  t[idx] = ((i == j) ? c : 0.f) - x[idx];
}

// depthwise 33-tap conv along tokens of v (read in place from qkv v-section)
__global__ void nys_res_conv(const float* __restrict__ qkv,
                             const float* __restrict__ rw,
                             float* __restrict__ out) {
  int idx = blockIdx.x * blockDim.x + threadIdx.x;
  if (idx >= 16 * 8 * 512 * 64) return;
  int d = idx & 63;
  int i = (idx >> 6) & 511;
  int h = (idx >> 15) & 7;
  int b = idx >> 18;
  const float* rwh = rw + h * 33;
  float s = 0.f;
  for (int t = 0; t < 33; ++t) {
    int ii = i + t - 16;
    if (ii >= 0 && ii < 512)
      s += qkv[((size_t)b * 512 + ii) * 1536 + 1024 + h * 64 + d] * rwh[t];
  }
  out[idx] = s;
}

// (B,H,N,64)+(B,H,N,64) -> (B,N,H*64)
__global__ void merge_heads_add(const float* __restrict__ nout,
                                const float* __restrict__ resv,
                                float* __restrict__ out) {
  int idx = blockIdx.x * blockDim.x + threadIdx.x;
  if (idx >= 16 * 512 * 512) return;
  int d = idx & 63;
  int h = (idx >> 6) & 7;
  int i = (idx >> 9) & 511;
  int b = idx >> 18;
  int src = (b << 18) | (h << 15) | (i << 6) | d;
  out[idx] = nout[src] + resv[src];
}

__global__ void copy_cls(const float* __restrict__ cls, float* __restrict__ xt,
                         int B) {
  int idx = blockIdx.x * blockDim.x + threadIdx.x;
  if (idx >= B * 512) return;
  int d = idx & 511, b = idx >> 9;
  xt[(size_t)b * 512 * 512 + d] = cls[d];
}

// ---------------------------------------------------------------------------
// Orchestration
// ---------------------------------------------------------------------------
extern "C" void kernel_launch(void* const* d_in, const int* in_sizes, int n_in,
                              void* d_out, int out_size, void* d_ws,
                              size_t ws_size, hipStream_t stream) {
  (void)in_sizes;
  (void)n_in;
  (void)out_size;
  (void)ws_size;
  auto F = [&](int i) { return (const float*)d_in[i]; };
  // Flat input map (insertion-order, depth-first):
  // 0 imu_data 1 thm_data 2 tof_data
  // sensor block (14): fe_w fe_b proj_w proj_b pos1_w pos1_b pos2_w pos2_b
  //                    wave1 wave2 wave3 wave1_ wave2_ wave3_
  // imu@3 tof@17 thm@31
  // fusion@45: it{in_w,in_b,out_w,out_b} ih@49 th@53, 57 fuse_w 58 fuse_b
  //            59 norm_g 60 norm_b
  // 61 cls_token
  // layer1@62: norm_g norm_b qkv_w out_w out_b res_w ; layer2@68
  // 74 norm_g 75 norm_b 76 fc2_w1 77 fc2_b1 78 fc2_w2 79 fc2_b2
  // 80 aux_w1 81 aux_b1 82 aux_w2 83 aux_b2

  const int Bsz = 16, L = 511, LF = 510, D = 512, Hh = 8, DHd = 64, NT = 512,
            LMm = 256;
  float* W = (float*)d_ws;
  size_t off = 0;
  auto alloc = [&](size_t n) {
    size_t o = off;
    off += (n + 63) & ~(size_t)63;
    return o;
  };
  float* convbuf = W + alloc((size_t)Bsz * L * 640);
  float* posbuf = W + alloc((size_t)Bsz * LF * D);
  float* wkbuf = W + alloc(512 * 19);
  float* sx[3];
  for (int i = 0; i < 3; ++i) sx[i] = W + alloc((size_t)Bsz * L * D);
  float* qbuf = W + alloc((size_t)Bsz * L * D);
  float* kvbuf = W + alloc((size_t)Bsz * L * 1024);
  float* attbuf = W + alloc((size_t)Bsz * Hh * L * L);
  float* obuf = W + alloc((size_t)Bsz * L * D);
  float* concat = W + alloc((size_t)Bsz * L * 1536);
  float* xtok = W + alloc((size_t)Bsz * NT * D);
  float* lnbuf = W + alloc((size_t)Bsz * NT * D);
  float* qkvbuf = W + alloc((size_t)Bsz * NT * 1536);
  float* qlbuf = W + alloc((size_t)128 * LMm * DHd);
  float* klbuf = W + alloc((size_t)128 * LMm * DHd);
  float* a1 = W + alloc((size_t)128 * NT * LMm);
  float* a2 = W + alloc((size_t)128 * LMm * LMm);
  float* a3 = W + alloc((size_t)128 * LMm * NT);
  float* zA = W + alloc((size_t)128 * LMm * LMm);
  float* zB = W + alloc((size_t)128 * LMm * LMm);
  float* xzb = W + alloc((size_t)128 * LMm * LMm);
  float* t1b = W + alloc((size_t)128 * LMm * LMm);
  float* t2b = W + alloc((size_t)128 * LMm * LMm);
  float* a3v = W + alloc((size_t)128 * LMm * DHd);
  float* pzb = W + alloc((size_t)128 * LMm * DHd);
  float* noutb = W + alloc((size_t)128 * NT * DHd);
  float* resvb = W + alloc((size_t)128 * NT * DHd);
  float* mrg = W + alloc((size_t)Bsz * NT * D);
  float* redR = W + alloc(32768);
  float* redC = W + alloc(32768);
  float* scb = W + alloc(64);
  float* x0 = W + alloc((size_t)Bsz * D);
  float* hb = W + alloc((size_t)Bsz * D);

  auto gemm = [&](const float* A, int lda, long long sAo, long long sAi,
                  const float* Bp, int ldb, long long sBo, long long sBi,
                  bool tB, float* Cp, int ldc, long long sCo, long long sCi,
                  const float* bias, const float* resid, int M, int N, int K,
                  int nb, int inner, float alpha, float diag, int relu) {
    dim3 g((N + 127) / 128, (M + 63) / 64, nb), b(256);
    if (tB)
      gemm_wmma<1><<<g, b, 0, stream>>>(A, lda, sAo, sAi, Bp, ldb, sBo, sBi, Cp,
                                        ldc, sCo, sCi, bias, resid, M, N, K,
                                        inner, alpha, diag, relu);
    else
      gemm_wmma<0><<<g, b, 0, stream>>>(A, lda, sAo, sAi, Bp, ldb, sBo, sBi, Cp,
                                        ldc, sCo, sCi, bias, resid, M, N, K,
                                        inner, alpha, diag, relu);
  };
  auto softmax = [&](float* p, long long rows, int len) {
    softmax_rows<<<dim3((unsigned)((rows + 7) / 8)), dim3(256), 0, stream>>>(
        p, rows, len);
  };
  auto LN = [&](const float* x, const float* resid, const float* g,
                const float* bp, float* y, int rows, int rpb, long long xbs,
                long long xrs, long long ybs, long long yrs) {
    layer_norm_k<<<dim3((rows + 7) / 8), dim3(256), 0, stream>>>(
        x, resid, g, bp, y, rows, rpb, xbs, xrs, ybs, yrs);
  };

  // ---- per-sensor feature extraction + projection + 2x wavelet encoding ----
  struct SP {
    const float* data;
    int S, Cin, pb;
  };
  SP sps[3] = {{F(0), 1, 7, 3}, {F(2), 5, 64, 17}, {F(1), 5, 1, 31}};
  for (int si = 0; si < 3; ++si) {
    SP sp = sps[si];
    int OCrow = sp.S * 128;
    int total = Bsz * L * OCrow;
    fe_conv_relu<<<dim3((total + 255) / 256), dim3(256), 0, stream>>>(
        sp.data, F(sp.pb + 0), F(sp.pb + 1), convbuf, Bsz, sp.S, L, sp.Cin,
        OCrow);
    gemm(convbuf, OCrow, 0, 0, F(sp.pb + 2), OCrow, 0, 0, true, sx[si], D, 0, 0,
         F(sp.pb + 3), nullptr, Bsz * L, D, OCrow, 1, 1, 1.f, 0.f, 0);
    for (int e = 0; e < 2; ++e) {
      int wv = sp.pb + 8 + e * 3;  // wave1..3 / wave1_..3_
      int pw = sp.pb + 4 + e * 2;  // pos{1,2}_{w,b}
      build_wavelet<<<dim3((512 * 19 + 255) / 256), dim3(256), 0, stream>>>(
          F(wv), F(wv + 1), F(wv + 2), wkbuf);
      int tp = Bsz * LF * D;
      wavelet_pos<<<dim3((tp + 255) / 256), dim3(256), 0, stream>>>(
          sx[si], wkbuf, posbuf, Bsz);
      // feat(tokens 1..510) += pos @ posW^T + posB   (residual in-place)
      gemm(posbuf, D, (long long)LF * D, 0, F(pw), D, 0, 0, true, sx[si] + D, D,
           (long long)L * D, 0, F(pw + 1), sx[si] + D, LF, D, D, Bsz, 1, 1.f,
           0.f, 0);
    }
  }

  // ---- fusion: 3x cross-MHA -> concat -> fuse -> LN(+imu residual) ----
  struct MH {
    const float* q;
    const float* kv;
    int wb, col;
  };
  MH ms[3] = {{sx[0], sx[1], 45, 0},
              {sx[0], sx[2], 49, 512},
              {sx[1], sx[2], 53, 1024}};
  for (int mi = 0; mi < 3; ++mi) {
    MH m = ms[mi];
    const float* inw = F(m.wb + 0);
    const float* inb = F(m.wb + 1);
    gemm(m.q, D, 0, 0, inw, D, 0, 0, true, qbuf, D, 0, 0, inb, nullptr, Bsz * L,
         D, D, 1, 1, 1.f, 0.f, 0);
    gemm(m.kv, D, 0, 0, inw + (size_t)512 * 512, D, 0, 0, true, kvbuf, 1024, 0,
         0, inb + 512, nullptr, Bsz * L, 1024, D, 1, 1, 1.f, 0.f, 0);
    // scores = 0.125 * q @ k^T   (head-strided views, batch z=(b,h))
    gemm(qbuf, D, (long long)L * D, 64, kvbuf, 1024, (long long)L * 1024, 64,
         true, attbuf, L, (long long)Hh * L * L, (long long)L * L, nullptr,
         nullptr, L, L, DHd, Bsz * Hh, Hh, 0.125f, 0.f, 0);
    softmax(attbuf, (long long)Bsz * Hh * L, L);
    // o = att @ v -> (B,L,H*64)
    gemm(attbuf, L, (long long)Hh * L * L, (long long)L * L, kvbuf + 512, 1024,
         (long long)L * 1024, 64, false, obuf, D, (long long)L * D, 64, nullptr,
         nullptr, L, DHd, L, Bsz * Hh, Hh, 1.f, 0.f, 0);
    gemm(obuf, D, 0, 0, F(m.wb + 2), D, 0, 0, true, concat + m.col, 1536, 0, 0,
         F(m.wb + 3), nullptr, Bsz * L, D, D, 1, 1, 1.f, 0.f, 0);
  }
  gemm(concat, 1536, 0, 0, F(57), 1536, 0, 0, true, qbuf, D, 0, 0, F(58),
       nullptr, Bsz * L, D, 1536, 1, 1, 1.f, 0.f, 0);
  copy_cls<<<dim3((Bsz * D + 255) / 256), dim3(256), 0, stream>>>(F(61), xtok,
                                                                  Bsz);
  LN(qbuf, sx[0], F(59), F(60), xtok + D, Bsz * L, L, (long long)L * D, D,
     (long long)NT * D, D);

  // ---- 2x Nystrom transformer layers ----
  for (int li = 0; li < 2; ++li) {
    int pb = 62 + li * 6;
    LN(xtok, nullptr, F(pb + 0), F(pb + 1), lnbuf, Bsz * NT, Bsz * NT, 0, D, 0,
       D);
    gemm(lnbuf, D, 0, 0, F(pb + 2), D, 0, 0, true, qkvbuf, 1536, 0, 0, nullptr,
         nullptr, Bsz * NT, 1536, D, 1, 1, 1.f, 0.f, 0);
    int tl = 128 * LMm * DHd;
    landmark_mean<<<dim3((tl + 255) / 256), dim3(256), 0, stream>>>(qkvbuf,
                                                                    qlbuf, 0);
    landmark_mean<<<dim3((tl + 255) / 256), dim3(256), 0, stream>>>(qkvbuf,
                                                                    klbuf, 512);
    // a1 = softmax(0.125 * q @ kl^T)
    gemm(qkvbuf, 1536, (long long)NT * 1536, 64, klbuf, 64,
         (long long)Hh * LMm * DHd, (long long)LMm * DHd, true, a1, LMm,
         (long long)Hh * NT * LMm, (long long)NT * LMm, nullptr, nullptr, NT,
         LMm, DHd, Bsz * Hh, Hh, 0.125f, 0.f, 0);
    // a2 = softmax(0.125 * ql @ kl^T)
    gemm(qlbuf, 64, (long long)LMm * DHd, 0, klbuf, 64, (long long)LMm * DHd, 0,
         true, a2, LMm, (long long)LMm * LMm, 0, nullptr, nullptr, LMm, LMm,
         DHd, Bsz * Hh, 1, 0.125f, 0.f, 0);
    // a3 = softmax(0.125 * ql @ k^T)
    gemm(qlbuf, 64, (long long)Hh * LMm * DHd, (long long)LMm * DHd,
         qkvbuf + 512, 1536, (long long)NT * 1536, 64, true, a3, NT,
         (long long)Hh * LMm * NT, (long long)LMm * NT, nullptr, nullptr, LMm,
         NT, DHd, Bsz * Hh, Hh, 0.125f, 0.f, 0);
    softmax(a1, (long long)128 * NT, LMm);
    softmax(a2, (long long)128 * LMm, LMm);
    softmax(a3, (long long)128 * LMm, NT);
    // Moore-Penrose pinv(a2), 6 Newton iterations
    abs_rowsum<<<dim3(128), dim3(256), 0, stream>>>(a2, redR);
    abs_colsum<<<dim3(128), dim3(256), 0, stream>>>(a2, redC);
    pinv_scale<<<dim3(1), dim3(256), 0, stream>>>(redR, redC, scb, 32768);
    transpose_scale<<<dim3(32768), dim3(256), 0, stream>>>(a2, scb, zA);
    float* zc = zA;
    float* zn = zB;
    long long sq = (long long)LMm * LMm;
    for (int it = 0; it < 6; ++it) {
      gemm(a2, LMm, sq, 0, zc, LMm, sq, 0, false, xzb, LMm, sq, 0, nullptr,
           nullptr, LMm, LMm, LMm, 128, 1, 1.f, 0.f, 0);
      ident_sub<<<dim3(32768), dim3(256), 0, stream>>>(xzb, t1b, 7.f);
      gemm(xzb, LMm, sq, 0, t1b, LMm, sq, 0, false, t2b, LMm, sq, 0, nullptr,
           nullptr, LMm, LMm, LMm, 128, 1, 1.f, 0.f, 0);
      ident_sub<<<dim3(32768), dim3(256), 0, stream>>>(t2b, t1b, 15.f);
      gemm(xzb, LMm, sq, 0, t1b, LMm, sq, 0, false, t2b, LMm, sq, 0, nullptr,
           nullptr, LMm, LMm, LMm, 128, 1, 1.f, 0.f, 0);
      ident_sub<<<dim3(32768), dim3(256), 0, stream>>>(t2b, t1b, 13.f);
      gemm(zc, LMm, sq, 0, t1b, LMm, sq, 0, false, zn, LMm, sq, 0, nullptr,
           nullptr, LMm, LMm, LMm, 128, 1, 0.25f, 0.f, 0);
      float* tmp = zc;
      zc = zn;
      zn = tmp;
    }
    // a3v = a3 @ v ; pz = pinv @ a3v ; nout = a1 @ pz
    gemm(a3, NT, (long long)Hh * LMm * NT, (long long)LMm * NT, qkvbuf + 1024,
         1536, (long long)NT * 1536, 64, false, a3v, DHd,
         (long long)Hh * LMm * DHd, (long long)LMm * DHd, nullptr, nullptr, LMm,
         DHd, NT, Bsz * Hh, Hh, 1.f, 0.f, 0);
    gemm(zc, LMm, sq, 0, a3v, DHd, (long long)LMm * DHd, 0, false, pzb, DHd,
         (long long)LMm * DHd, 0, nullptr, nullptr, LMm, DHd, LMm, 128, 1, 1.f,
         0.f, 0);
    gemm(a1, LMm, (long long)NT * LMm, 0, pzb, DHd, (long long)LMm * DHd, 0,
         false, noutb, DHd, (long long)NT * DHd, 0, nullptr, nullptr, NT, DHd,
         LMm, 128, 1, 1.f, 0.f, 0);
    nys_res_conv<<<dim3(16384), dim3(256), 0, stream>>>(qkvbuf, F(pb + 5),
                                                        resvb);
    merge_heads_add<<<dim3(16384), dim3(256), 0, stream>>>(noutb, resvb, mrg);
    // xtok += mrg @ out_w^T + out_b
    gemm(mrg, D, 0, 0, F(pb + 3), D, 0, 0, true, xtok, D, 0, 0, F(pb + 4), xtok,
         Bsz * NT, D, D, 1, 1, 1.f, 0.f, 0);
  }

  // ---- final LN (token 0) + classifier heads ----
  LN(xtok, nullptr, F(74), F(75), x0, Bsz, Bsz, 0, (long long)NT * D, 0, D);
  float* out = (float*)d_out;
  gemm(x0, D, 0, 0, F(76), D, 0, 0, true, hb, D, 0, 0, F(77), nullptr, Bsz, D,
       D, 1, 1, 1.f, 0.f, 1);
  gemm(hb, D, 0, 0, F(78), D, 0, 0, true, out, 18, 0, 0, F(79), nullptr, Bsz,
       18, D, 1, 1, 1.f, 0.f, 0);
  gemm(x0, D, 0, 0, F(80), D, 0, 0, true, hb, D, 0, 0, F(81), nullptr, Bsz, D,
       D, 1, 1, 1.f, 0.f, 1);
  gemm(hb, D, 0, 0, F(82), D, 0, 0, true, out + Bsz * 18, 2, 0, 0, F(83),
       nullptr, Bsz, 2, D, 1, 1, 1.f, 0.f, 0);
}